// SlidingFFAgent_8641474200160
// MI455X (gfx1250) — compile-verified
//
#include <hip/hip_runtime.h>

// ---- problem dims (match reference) ----
#define BSZ  32
#define TSQ  256
#define NAG  16
#define OSZ  512     // encoder input dim
#define HID  1024
#define DD   128     // encoded frame dim
#define WINW 8
#define OUTD 512
#define DECD 512
#define NACT 30

#define SEQ_ROWS (TSQ + WINW - 1)   // 263 frames resident in LDS
// LDS row strides padded so (stride % 64banks) != 0 -> A-frag reads hit 16 banks;
// strides stay multiples of 8B so ds_load_b64 A-fragments remain naturally aligned.
#define SEQ_LD 132                  // frames: 128 cols + 4 pad (528 B)
#define A_LD   516                  // 64x512 staging: 512 + 4 pad (2064 B)
#define C1_LD  132                  // 64x128 chunk: 128 + 4 pad

typedef __attribute__((ext_vector_type(2))) float v2f;
typedef __attribute__((ext_vector_type(8))) float v8f;

static __device__ __forceinline__ v8f vzero8() {
  v8f v;
#pragma unroll
  for (int i = 0; i < 8; ++i) v[i] = 0.0f;
  return v;
}

// D = A(16x4) * B(4x16) + C  on one wave32
static __device__ __forceinline__ v8f wmma4(v2f a, v2f b, v8f c) {
  return __builtin_amdgcn_wmma_f32_16x16x4_f32(false, a, false, b, (short)0, c,
                                               false, false);
}

// A fragment 16x4 from LDS: lanes 0-15 rows M0..M0+15 (k,k+1); lanes 16-31 (k+2,k+3)
static __device__ __forceinline__ v2f load_a(const float* base, int ld, int row0,
                                             int k, int lane) {
  const int r  = row0 + (lane & 15);
  const int kk = k + ((lane >> 4) << 1);
  const float* p = base + r * ld + kk;
  v2f f; f[0] = p[0]; f[1] = p[1];
  return f;
}

// A fragment for the sliding-window GEMM: wins[r][k] == seq[r + k/128][k%128]
static __device__ __forceinline__ v2f load_a_seq(const float* sseq, int row0,
                                                 int k, int lane) {
  const int r  = row0 + (lane & 15);
  const int kk = k + ((lane >> 4) << 1);       // kk,kk+1 never cross a frame
  const float* p = sseq + (r + (kk >> 7)) * SEQ_LD + (kk & 127);
  v2f f; f[0] = p[0]; f[1] = p[1];
  return f;
}

// B fragment 4x16 from global row-major (K x N): lanes 0-15 cols (rows k,k+1),
// lanes 16-31 (rows k+2,k+3)
static __device__ __forceinline__ v2f load_b(const float* Wm, int ldw, int k,
                                             int col0, int lane) {
  const int c  = col0 + (lane & 15);
  const int kk = k + ((lane >> 4) << 1);
  v2f f;
  f[0] = Wm[(size_t)kk * ldw + c];
  f[1] = Wm[(size_t)(kk + 1) * ldw + c];
  return f;
}

__launch_bounds__(256, 1)
__global__ void sliding_ff_agent_fused(
    const float* __restrict__ gin,   // (BS,TS,NA,OS)
    const float* __restrict__ ghid,  // (BS,NA,(W-1)*D)
    const float* __restrict__ w1, const float* __restrict__ b1,
    const float* __restrict__ w2, const float* __restrict__ b2,
    const float* __restrict__ w3, const float* __restrict__ b3,
    const float* __restrict__ w4, const float* __restrict__ b4,
    const float* __restrict__ w5, const float* __restrict__ b5,
    float* __restrict__ qout,        // (BS,TS,NA,NACT)
    float* __restrict__ wout)        // (BS,NA,(W-1)*D)
{
  __shared__ float s_seq[SEQ_ROWS * SEQ_LD]; // 138,864 B: full frame sequence
  __shared__ float s_a[64 * A_LD];           // 132,096 B: input tile / h3 tile
  __shared__ float s_c1[64 * C1_LD];         //  33,792 B: 64x128 chunk buffer

  const int b    = blockIdx.x >> 4;   // batch
  const int a    = blockIdx.x & 15;   // agent
  const int tid  = threadIdx.x;
  const int lane = tid & 31;
  const int wv   = tid >> 5;          // wave 0..7
  const int lrow = lane & 15;
  const int hsel = (lane >> 4) << 3;  // C/D row offset: 0 or 8

  // ---- hidden_state -> seq rows 0..6 ----
  {
    const float* src = ghid + (size_t)(b * NAG + a) * ((WINW - 1) * DD);
    for (int i = tid; i < (WINW - 1) * DD; i += 256)
      s_seq[(i >> 7) * SEQ_LD + (i & 127)] = src[i];
  }

  // =========================== ENCODER ===========================
  // per 64-row tile: L1 (K=512,N=1024 in 8 chunks of 128) fused with L2 (K=1024,N=128)
  for (int m0 = 0; m0 < TSQ; m0 += 64) {
    // stage inputs tile (64 x 512) -> s_a
    for (int idx = tid; idx < 64 * 128; idx += 256) {
      const int r = idx >> 7, g = idx & 127;
      const float4* srow =
          (const float4*)(gin + (size_t)((b * TSQ + (m0 + r)) * NAG + a) * OSZ);
      *(float4*)(s_a + r * A_LD + g * 4) = srow[g];
    }
    __syncthreads();

    // prefetch NEXT input tile into cache while this tile computes:
    // 64 rows x 2KB = 1024 x 128B cachelines; 256 threads x 4 lines each.
    if (m0 + 64 < TSQ) {
#pragma unroll
      for (int i = 0; i < 4; ++i) {
        const int idx  = tid + 256 * i;        // 0..1023
        const int r    = idx >> 4;             // row 0..63
        const int line = idx & 15;             // 128B line within row
        const float* p =
            gin + (size_t)((b * TSQ + (m0 + 64 + r)) * NAG + a) * OSZ + line * 32;
        __builtin_prefetch(p, 0, 1);           // -> global_prefetch_b8
      }
    }

    v8f c2[4];                       // this wave's 64 x 16 slice of L2 output
#pragma unroll
    for (int i = 0; i < 4; ++i) c2[i] = vzero8();

    for (int j = 0; j < 8; ++j) {    // HID chunks of 128
      // --- GEMM1: s_a(64x512) @ w1[:, chunk] -> relu -> s_c1 (64x128) ---
      v8f acc1[4];
#pragma unroll
      for (int i = 0; i < 4; ++i) acc1[i] = vzero8();
      const int ncol1 = j * 128 + wv * 16;
#pragma unroll 2
      for (int k = 0; k < OSZ; k += 4) {
        const v2f bf = load_b(w1, HID, k, ncol1, lane);
#pragma unroll
        for (int mt = 0; mt < 4; ++mt)
          acc1[mt] = wmma4(load_a(s_a, A_LD, mt * 16, k, lane), bf, acc1[mt]);
      }
      const float bias1 = b1[ncol1 + lrow];
#pragma unroll
      for (int mt = 0; mt < 4; ++mt) {
        const int rbase = mt * 16 + hsel;
#pragma unroll
        for (int g = 0; g < 8; ++g) {
          float v = acc1[mt][g] + bias1;
          s_c1[(rbase + g) * C1_LD + wv * 16 + lrow] = v > 0.f ? v : 0.f;
        }
      }
      __syncthreads();

      // --- GEMM2: s_c1(64x128) @ w2[chunk rows, :] -> accumulate c2 ---
      const float* w2c = w2 + (size_t)j * 128 * DD;
#pragma unroll 2
      for (int k = 0; k < 128; k += 4) {
        const v2f bf = load_b(w2c, DD, k, wv * 16, lane);
#pragma unroll
        for (int mt = 0; mt < 4; ++mt)
          c2[mt] = wmma4(load_a(s_c1, C1_LD, mt * 16, k, lane), bf, c2[mt]);
      }
      __syncthreads();
    }

    // bias + relu -> seq rows 7+m0 ..
    const float bias2 = b2[wv * 16 + lrow];
#pragma unroll
    for (int mt = 0; mt < 4; ++mt) {
      const int rbase = (WINW - 1) + m0 + mt * 16 + hsel;
#pragma unroll
      for (int g = 0; g < 8; ++g) {
        float v = c2[mt][g] + bias2;
        s_seq[(rbase + g) * SEQ_LD + wv * 16 + lrow] = v > 0.f ? v : 0.f;
      }
    }
    __syncthreads();
  }

  // =========================== DECODER ===========================
  // per 64-row tile: L3 (sliding window GEMM, K=1024) -> s_a,
  // then L4 (K=512, N=512 in 4 chunks) fused with L5 (K=512, N=30)
  for (int m0 = 0; m0 < TSQ; m0 += 64) {
    // --- L3: wins(64x1024) @ w3 -> relu -> s_a (64x512) ---
    for (int pass = 0; pass < 2; ++pass) {
      const int col0 = (wv + pass * 8) * 32;
      v8f acc[2][4];
#pragma unroll
      for (int nt = 0; nt < 2; ++nt)
#pragma unroll
        for (int mt = 0; mt < 4; ++mt) acc[nt][mt] = vzero8();

#pragma unroll 2
      for (int k = 0; k < WINW * DD; k += 4) {
        const v2f bf0 = load_b(w3, OUTD, k, col0, lane);
        const v2f bf1 = load_b(w3, OUTD, k, col0 + 16, lane);
#pragma unroll
        for (int mt = 0; mt < 4; ++mt) {
          const v2f af = load_a_seq(s_seq, m0 + mt * 16, k, lane);
          acc[0][mt] = wmma4(af, bf0, acc[0][mt]);
          acc[1][mt] = wmma4(af, bf1, acc[1][mt]);
        }
      }
#pragma unroll
      for (int nt = 0; nt < 2; ++nt) {
        const float bias3 = b3[col0 + nt * 16 + lrow];
#pragma unroll
        for (int mt = 0; mt < 4; ++mt) {
          const int rbase = mt * 16 + hsel;
#pragma unroll
          for (int g = 0; g < 8; ++g) {
            float v = acc[nt][mt][g] + bias3;
            s_a[(rbase + g) * A_LD + col0 + nt * 16 + lrow] = v > 0.f ? v : 0.f;
          }
        }
      }
    }
    __syncthreads();

    // --- L4 + L5 fused over DEC chunks of 128 ---
    v8f qacc = vzero8();            // wave's 16x16 q tile
    const int qmt = wv & 3;         // m-tile 0..3
    const int qcol = (wv >> 2) * 16 + lrow;   // 0..31, valid if < NACT
    const bool cvalid = qcol < NACT;

    for (int j = 0; j < 4; ++j) {
      // GEMM4: s_a(64x512) @ w4[:, chunk] -> relu -> s_c1 (64x128)
      v8f acc4[4];
#pragma unroll
      for (int i = 0; i < 4; ++i) acc4[i] = vzero8();
      const int ncol4 = j * 128 + wv * 16;
#pragma unroll 2
      for (int k = 0; k < OUTD; k += 4) {
        const v2f bf = load_b(w4, DECD, k, ncol4, lane);
#pragma unroll
        for (int mt = 0; mt < 4; ++mt)
          acc4[mt] = wmma4(load_a(s_a, A_LD, mt * 16, k, lane), bf, acc4[mt]);
      }
      const float bias4 = b4[ncol4 + lrow];
#pragma unroll
      for (int mt = 0; mt < 4; ++mt) {
        const int rbase = mt * 16 + hsel;
#pragma unroll
        for (int g = 0; g < 8; ++g) {
          float v = acc4[mt][g] + bias4;
          s_c1[(rbase + g) * C1_LD + wv * 16 + lrow] = v > 0.f ? v : 0.f;
        }
      }
      __syncthreads();

      // GEMM5: s_c1(64x128) @ w5[chunk rows, :30] -> qacc (zero-fill cols >=30;
      // EXEC stays all-ones around wmma)
#pragma unroll 2
      for (int k = 0; k < 128; k += 4) {
        const int kk = j * 128 + k + ((lane >> 4) << 1);
        v2f bf;
        bf[0] = cvalid ? w5[(size_t)kk * NACT + qcol] : 0.f;
        bf[1] = cvalid ? w5[(size_t)(kk + 1) * NACT + qcol] : 0.f;
        qacc = wmma4(load_a(s_c1, C1_LD, qmt * 16, k, lane), bf, qacc);
      }
      __syncthreads();
    }

    // store q tile
    if (cvalid) {
      const float bias5 = b5[qcol];
      const int rbase = m0 + qmt * 16 + hsel;
#pragma unroll
      for (int g = 0; g < 8; ++g) {
        const int t = rbase + g;
        qout[(size_t)((b * TSQ + t) * NAG + a) * NACT + qcol] = qacc[g] + bias5;
      }
    }
  }

  // ---- carried window: seq rows 256..262 -> wout ----
  {
    float* dst = wout + (size_t)(b * NAG + a) * ((WINW - 1) * DD);
    for (int i = tid; i < (WINW - 1) * DD; i += 256)
      dst[i] = s_seq[(TSQ + (i >> 7)) * SEQ_LD + (i & 127)];
  }
}

extern "C" void kernel_launch(void* const* d_in, const int* in_sizes, int n_in,
                              void* d_out, int out_size, void* d_ws, size_t ws_size,
                              hipStream_t stream) {
  (void)in_sizes; (void)n_in; (void)out_size; (void)d_ws; (void)ws_size;
  const float* gin  = (const float*)d_in[0];
  const float* ghid = (const float*)d_in[1];
  const float* w1 = (const float*)d_in[2];
  const float* b1 = (const float*)d_in[3];
  const float* w2 = (const float*)d_in[4];
  const float* b2 = (const float*)d_in[5];
  const float* w3 = (const float*)d_in[6];
  const float* b3 = (const float*)d_in[7];
  const float* w4 = (const float*)d_in[8];
  const float* b4 = (const float*)d_in[9];
  const float* w5 = (const float*)d_in[10];
  const float* b5 = (const float*)d_in[11];
  float* qout = (float*)d_out;
  float* wout = qout + (size_t)BSZ * TSQ * NAG * NACT;

  sliding_ff_agent_fused<<<dim3(BSZ * NAG), dim3(256), 0, stream>>>(
      gin, ghid, w1, b1, w2, b2, w3, b3, w4, b4, w5, b5, qout, wout);
}